// ReprojectionLayer_41472204210545
// MI455X (gfx1250) — compile-verified
//
#include <hip/hip_runtime.h>

// ---------------------------------------------------------------------------
// ReprojectionLayer: gather 12-camera heatmap values per voxel and average.
//   heatmaps   : [B=2, C=12, J=23, 512, 512]  f32
//   center     : [B, 3]                        f32
//   reproLookup: [C=12, 160, 160, 160, 2]      i32
//   out        : [B, 23, 104, 104, 104]        f32
// Memory-bound gather; cam-outer / joint-inner so one 8B index feeds 23
// gathers; async LDS double-buffer (ASYNCcnt) hides LUT latency; NT hints
// keep streaming traffic out of the hot L2 set.
// ---------------------------------------------------------------------------

#define NUM_CAMS   12
#define NUM_JOINTS 23
#define GSZ        104
#define HALFG      52
#define LGRID      160
#define HM_W       512
#define HM_HW      (512 * 512)
#define G3         (GSZ * GSZ * GSZ)          // 1,124,864  (== 4394 * 256)
#define L3         ((size_t)LGRID * LGRID * LGRID)
#define GRID_OFF   (-160.0f)
#define GRID_SP    2.0f

#if defined(__has_builtin)
#if __has_builtin(__builtin_amdgcn_global_load_async_to_lds_b64) && \
    __has_builtin(__builtin_amdgcn_s_wait_asynccnt)
#define HAVE_ASYNC_LDS 1
#endif
#endif
#ifndef HAVE_ASYNC_LDS
#define HAVE_ASYNC_LDS 0
#endif

typedef int v2i __attribute__((ext_vector_type(2)));       // builtin data type
typedef __attribute__((address_space(3))) v2i LdsV2i;      // builtin dst type

__global__ __launch_bounds__(256) void repro_gather_mean_kernel(
    const float* __restrict__ hm,      // [B, C, J, 512, 512]
    const float* __restrict__ center,  // [B, 3]
    const int* __restrict__ lut,       // [C, L, L, L, 2]
    float* __restrict__ out)           // [B, J, G, G, G]
{
  const int tid = threadIdx.x;
  const int vox = blockIdx.x * 256 + tid;
  const int b   = blockIdx.y;
  if (vox >= G3) return;

  // voxel coords: vox = x*G^2 + y*G + z (matches reference flattening)
  const int z = vox % GSZ;
  const int t = vox / GSZ;
  const int y = t % GSZ;
  const int x = t / GSZ;

  // center_index = trunc((center - offset) / spacing); slice start = idx - 52
  const int sx = (int)((center[b * 3 + 0] - GRID_OFF) / GRID_SP) - HALFG;
  const int sy = (int)((center[b * 3 + 1] - GRID_OFF) / GRID_SP) - HALFG;
  const int sz = (int)((center[b * 3 + 2] - GRID_OFF) / GRID_SP) - HALFG;

  // pair index into lut viewed as int2 pairs: (gx*L + gy)*L + gz  (+ c*L^3)
  const size_t pairBase =
      ((size_t)(sx + x) * LGRID + (size_t)(sy + y)) * LGRID + (size_t)(sz + z);

  const v2i* __restrict__ lutp = reinterpret_cast<const v2i*>(lut);

#if HAVE_ASYNC_LDS
  __shared__ v2i sIdx[2][256];
  // kick off cam 0's (u,v) pair into LDS (ASYNCcnt path)
  __builtin_amdgcn_global_load_async_to_lds_b64(
      (v2i*)(lutp + pairBase), (LdsV2i*)&sIdx[0][tid], 0, 0);
#endif

  float acc[NUM_JOINTS];
#pragma unroll
  for (int j = 0; j < NUM_JOINTS; ++j) acc[j] = 0.0f;

  const float* __restrict__ hmB =
      hm + (size_t)b * NUM_CAMS * NUM_JOINTS * HM_HW;

  for (int c = 0; c < NUM_CAMS; ++c) {
    v2i pair;
#if HAVE_ASYNC_LDS
    if (c + 1 < NUM_CAMS) {
      // double-buffer: stream cam c+1 indices while cam c gathers run
      __builtin_amdgcn_global_load_async_to_lds_b64(
          (v2i*)(lutp + (size_t)(c + 1) * L3 + pairBase),
          (LdsV2i*)&sIdx[(c + 1) & 1][tid], 0, 0);
      __builtin_amdgcn_s_wait_asynccnt(1);  // async loads complete in order
    } else {
      __builtin_amdgcn_s_wait_asynccnt(0);
    }
    pair = sIdx[c & 1][tid];
#else
    if (c + 1 < NUM_CAMS)
      __builtin_prefetch(lutp + (size_t)(c + 1) * L3 + pairBase, 0, 1);
    pair = __builtin_nontemporal_load(lutp + (size_t)c * L3 + pairBase);
#endif
    const int u    = pair.x;            // sub[...,0]
    const int v    = pair.y;            // sub[...,1]
    const int flat = v * HM_W + u;

    // 23 gathers share `flat`; joint stride (1 MB) is wave-uniform -> saddr
    const float* __restrict__ p =
        hmB + (size_t)c * NUM_JOINTS * HM_HW + (size_t)flat;
#pragma unroll
    for (int j = 0; j < NUM_JOINTS; ++j) acc[j] += p[(size_t)j * HM_HW];
  }

  // mean over cameras; NT stores: output is write-once streaming (207 MB)
  float* __restrict__ o = out + (size_t)b * NUM_JOINTS * G3 + vox;
#pragma unroll
  for (int j = 0; j < NUM_JOINTS; ++j)
    __builtin_nontemporal_store(acc[j] * (1.0f / 12.0f), o + (size_t)j * G3);
}

extern "C" void kernel_launch(void* const* d_in, const int* in_sizes, int n_in,
                              void* d_out, int out_size, void* d_ws,
                              size_t ws_size, hipStream_t stream) {
  const float* hm     = (const float*)d_in[0];
  const float* center = (const float*)d_in[1];
  const int* lut      = (const int*)d_in[2];
  float* out          = (float*)d_out;

  const int B = in_sizes[1] / 3;  // center is [B, 3]
  dim3 grid((G3 + 255) / 256, B);
  repro_gather_mean_kernel<<<grid, 256, 0, stream>>>(hm, center, lut, out);
}